// Net_32212254720595
// MI455X (gfx1250) — compile-verified
//
#include <hip/hip_runtime.h>
#include <stdint.h>

typedef __attribute__((ext_vector_type(2))) float    v2f;
typedef __attribute__((ext_vector_type(8))) float    v8f;
typedef __attribute__((ext_vector_type(4))) uint32_t v4u;
typedef __attribute__((ext_vector_type(4))) int      v4i;
typedef __attribute__((ext_vector_type(8))) int      v8i;

#define BATCH   128
#define T_STEPS 25
#define CIN     2
#define HIN     128
#define WIN     128
#define NCLS    10
#define KC      5
#define OH      124
#define OW      124
#define DECAY   0.05f
#define THRESH  1.0f

#define TILE_H  8
#define TILE_W  16
#define TILES_H 16   // ceil(124/8)
#define TILES_W 8    // ceil(124/16)
#define PH      (TILE_H + KC - 1)   // 12
#define PW      (TILE_W + KC - 1)   // 20
#define PATCH_N (CIN * PH * PW)     // 480
#define XPLANE  (CIN * HIN * WIN)   // elements per (b,t)

// ---- Tensor Data Mover availability / arity probing ----
#if defined(__has_builtin)
#  if __has_builtin(__builtin_amdgcn_tensor_load_to_lds)
#    define HAVE_TDM 1
#  endif
#endif
#ifndef HAVE_TDM
#  define HAVE_TDM 0
#endif
#if defined(__has_include)
#  if __has_include(<hip/amd_detail/amd_gfx1250_TDM.h>)
#    define TDM_SIX_ARGS 1   // therock-10.0 headers -> clang-23 6-arg builtin
#  else
#    define TDM_SIX_ARGS 0   // ROCm 7.2 clang-22 5-arg builtin
#  endif
#else
#  define TDM_SIX_ARGS 0
#endif

__global__ void zero_out_kernel(float* __restrict__ out, int n) {
  int i = blockIdx.x * blockDim.x + threadIdx.x;
  if (i < n) out[i] = 0.0f;
}

#if HAVE_TDM
// One 3D-tile TDM descriptor: tile 20(x) x 12(y) x 2(z) fp32 elements out of a
// tensor with row stride 128 and plane stride 128*128, zero-filling past the
// remaining extents (remw, remh) -> halo OOB handled by hardware.
__device__ __forceinline__ void tdm_load_patch(uint64_t ga, uint32_t lds_off,
                                               uint32_t remw, uint32_t remh) {
  v4u g0;
  g0[0] = 1u;                                   // count=1 (valid), no gather
  g0[1] = lds_off;                              // LDS byte address
  g0[2] = (uint32_t)ga;                         // global_addr[31:0]
  g0[3] = (uint32_t)(ga >> 32) | (2u << 30);    // global_addr[56:32] | type=2

  v8i g1;
  g1[0] = (int)(2u << 16);                      // data_size=4B, mask/flags=0
  g1[1] = (int)(remw << 16);                    // tensor_dim0[15:0]
  g1[2] = (int)(remh << 16);                    // dim0 hi=0 | tensor_dim1[15:0]
  g1[3] = (int)((uint32_t)PW << 16);            // dim1 hi=0 | tile_dim0=20
  g1[4] = (int)((uint32_t)PH | ((uint32_t)CIN << 16)); // tile_dim1=12, tile_dim2=2
  g1[5] = (int)WIN;                             // tensor_dim0_stride = 128
  g1[6] = (int)((uint32_t)(HIN * WIN) << 16);   // d0str hi=0 | d1stride lo16 = 16384
  g1[7] = 0;                                    // tensor_dim1_stride hi

  v4i g2; g2[0] = CIN; g2[1] = 0; g2[2] = 0; g2[3] = 0;  // tensor_dim2=2 (3D)
  v4i g3; g3[0] = 0;  g3[1] = 0; g3[2] = 0; g3[3] = 0;

#if TDM_SIX_ARGS
  v8i g4; g4[0]=0; g4[1]=0; g4[2]=0; g4[3]=0; g4[4]=0; g4[5]=0; g4[6]=0; g4[7]=0;
  __builtin_amdgcn_tensor_load_to_lds(g0, g1, g2, g3, g4, 0);
#else
  __builtin_amdgcn_tensor_load_to_lds(g0, g1, g2, g3, 0);
#endif
}
#endif

// Fused conv(5x5, 2->10) + leaky-integrate-fire over T steps + spatial score sum.
// One wave = 16-pixel row strip x 16 classes (10 real) via V_WMMA_F32_16X16X4_F32
// implicit GEMM; mem state lives in the D-fragment; input patches stream in via
// the Tensor Data Mover, double-buffered so DMA(t+1) overlaps compute(t).
__global__ __launch_bounds__(256)
void snn_conv_kernel(const float* __restrict__ x,
                     const float* __restrict__ cw,
                     const float* __restrict__ cb,
                     float* __restrict__ out) {
  __shared__ float patchbuf[2][PATCH_N];   // [c][pr][pc], pc fastest (TDM order)
  __shared__ float scw[8][16];

  const int tid   = threadIdx.x;
  const int wave  = tid >> 5;        // 0..7 -> row within tile
  const int lane  = tid & 31;
  const int n     = lane & 15;       // class column (B/C/D layout: N = lane%16)
  const int khalf = lane >> 4;       // 0/1  (A/B layout: which K pair)
  const int m     = lane & 15;       // A layout: M = lane%16

  int blk = blockIdx.x;
  const int b  = blk / (TILES_H * TILES_W);
  int r        = blk % (TILES_H * TILES_W);
  const int th = r / TILES_W, tw = r % TILES_W;
  const int h0 = th * TILE_H;
  const int w0 = tw * TILE_W;
  const int orow = h0 + wave;        // output row for this wave
  const bool rowv = (orow < OH);
  const bool nv   = (n < NCLS);

  // ---- B fragments (weights), resident across the whole kernel ----
  // B[k][n] = w[n][c][ki][kj] with k = c*25 + ki*5 + kj (k>=50 zero-padded)
  v2f bw[13];
  #pragma unroll
  for (int kb = 0; kb < 13; ++kb) {
    #pragma unroll
    for (int j = 0; j < 2; ++j) {
      int K = kb * 4 + khalf * 2 + j;
      float v = 0.0f;
      if (K < 50 && nv) {
        int c = K / 25, rem = K % 25;
        int ki = rem / 5, kj = rem % 5;
        v = cw[((n * CIN + c) * KC + ki) * KC + kj];
      }
      bw[kb][j] = v;
    }
  }
  const float bval = nv ? cb[n] : 0.0f;

  // D-fragment pixel/validity mask: slot v holds pixel M = v + 8*khalf
  float maskf[8];
  #pragma unroll
  for (int v = 0; v < 8; ++v) {
    int mm = v + 8 * khalf;
    maskf[v] = (rowv && nv && (w0 + mm) < OW) ? 1.0f : 0.0f;
  }

  v8f mem = {};          // mem0 = zeros
  float score = 0.0f;

  const float* xb = x + (size_t)b * T_STEPS * XPLANE;

#if HAVE_TDM
  const uint32_t remw = (uint32_t)(WIN - w0);
  const uint32_t remh = (uint32_t)(HIN - h0);
  const uint32_t lds0 = (uint32_t)(uintptr_t)&patchbuf[0][0];
  const uint32_t lds1 = (uint32_t)(uintptr_t)&patchbuf[1][0];
  const size_t   toff = (size_t)h0 * WIN + w0;   // tile origin within a plane
  if (wave == 0)
    tdm_load_patch((uint64_t)(uintptr_t)(xb + toff), lds0, remw, remh);
#endif

  for (int t = 0; t < T_STEPS; ++t) {
#if HAVE_TDM
    const float* patch = patchbuf[t & 1];
    if (wave == 0) {
      if (t + 1 < T_STEPS) {
        tdm_load_patch((uint64_t)(uintptr_t)(xb + (size_t)(t + 1) * XPLANE + toff),
                       ((t + 1) & 1) ? lds1 : lds0, remw, remh);
        __builtin_amdgcn_s_wait_tensorcnt((short)1);   // oldest (t) DMA complete
      } else {
        __builtin_amdgcn_s_wait_tensorcnt((short)0);
      }
    }
    __syncthreads();   // patch[t&1] visible to all waves
#else
    // Fallback: manual staging through VGPRs -> LDS
    const float* patch = patchbuf[0];
    const float* xt = xb + (size_t)t * XPLANE;
    __syncthreads();
    for (int i = tid; i < PATCH_N; i += 256) {
      int c = i / (PH * PW), rem2 = i % (PH * PW);
      int pr = rem2 / PW, pc = rem2 % PW;
      int gh = h0 + pr; if (gh > HIN - 1) gh = HIN - 1;
      int gw = w0 + pc; if (gw > WIN - 1) gw = WIN - 1;
      size_t goff = (size_t)c * HIN * WIN + (size_t)gh * WIN + gw;
      patchbuf[0][i] = xt[goff];
      if (t + 1 < T_STEPS)
        __builtin_prefetch(xt + (size_t)XPLANE + goff, 0, 1);
    }
    __syncthreads();
#endif

    // ---- implicit GEMM: conv[m][n] via 13x V_WMMA_F32_16X16X4_F32 ----
    v8f acc = {};
    #pragma unroll
    for (int kb = 0; kb < 13; ++kb) {
      v2f a;
      #pragma unroll
      for (int j = 0; j < 2; ++j) {
        int K = kb * 4 + khalf * 2 + j;
        float av = 0.0f;
        if (K < 50) {
          int c = K / 25, rem3 = K % 25;
          int ki = rem3 / 5, kj = rem3 % 5;
          av = patch[(c * PH + (wave + ki)) * PW + (m + kj)];
        }
        a[j] = av;
      }
      // 8 args: (neg_a, A, neg_b, B, c_mod, C, reuse_a, reuse_b)
      acc = __builtin_amdgcn_wmma_f32_16x16x4_f32(
          false, a, false, bw[kb], (short)0, acc, false, false);
    }

    // ---- LIF update + score accumulation (score = full spatial sum of mem) ----
    #pragma unroll
    for (int v = 0; v < 8; ++v) {
      float nm = mem[v] - DECAY + acc[v] + bval;
      score += nm * maskf[v];
      mem[v] = (nm >= THRESH) ? 0.0f : nm;
    }

    __syncthreads();   // all waves done with patch[t&1] before DMA(t+2) reuses it
  }

  // lane L and L^16 hold the same class n -> combine, then block reduce
  score += __shfl_xor(score, 16, 32);
  if (lane < 16) scw[wave][lane] = score;
  __syncthreads();
  if (tid < 16) {
    float s = 0.0f;
    #pragma unroll
    for (int wv = 0; wv < 8; ++wv) s += scw[wv][tid];
    if (tid < NCLS) atomicAdd(&out[b * NCLS + tid], s);
  }
}

extern "C" void kernel_launch(void* const* d_in, const int* in_sizes, int n_in,
                              void* d_out, int out_size, void* d_ws, size_t ws_size,
                              hipStream_t stream) {
  const float* x  = (const float*)d_in[0];
  const float* cw = (const float*)d_in[1];
  const float* cb = (const float*)d_in[2];
  float* out = (float*)d_out;
  (void)in_sizes; (void)n_in; (void)out_size; (void)d_ws; (void)ws_size;

  zero_out_kernel<<<(BATCH * NCLS + 255) / 256, 256, 0, stream>>>(out, BATCH * NCLS);

  dim3 grid(BATCH * TILES_H * TILES_W);   // 16384 blocks, 8 waves each
  snn_conv_kernel<<<grid, 256, 0, stream>>>(x, cw, cb, out);
}